// OhemCrossEntropy_69355131896499
// MI455X (gfx1250) — compile-verified
//
#include <hip/hip_runtime.h>
#include <hip/hip_bf16.h>
#include <math.h>

typedef __attribute__((ext_vector_type(2))) float v2f;
typedef __attribute__((ext_vector_type(8))) float v8f;

#define BN 8
#define CN 19
#define PH 128
#define PW 256
#define HH 512
#define WW 1024
#define NPIX (BN*HH*WW)          /* 4194304 */
#define MIN_KEPT_K 100000u
#define GRID_RED 1024

__device__ __forceinline__ unsigned orderKey(float f) {
  unsigned u = __float_as_uint(f);
  return (u & 0x80000000u) ? ~u : (u | 0x80000000u);
}
__device__ __forceinline__ float keyToFloat(unsigned k) {
  return (k & 0x80000000u) ? __uint_as_float(k & 0x7FFFFFFFu) : __uint_as_float(~k);
}

// ---------------------------------------------------------------------------
// Kernel A: bilinear 4x upsample (WMMA f32 16x16x4), log-softmax, CE, pred,
// high-16-bit radix histogram of pred keys, valid count.
// One wave32 = one 4x4 output tile. D[c][phase] = sum_k P[c][k] * W[phase][k].
// K = 9 corners padded to 12; padding handled by zero B-weights (A loads are
// unconditionally clamped to corner 8 -> fully branchless gather).
// ---------------------------------------------------------------------------
__global__ __launch_bounds__(256) void ohem_interp_ce(
    const float* __restrict__ score, const int* __restrict__ target,
    float* __restrict__ pred, float* __restrict__ loss,
    unsigned* __restrict__ hist, unsigned* __restrict__ meta)
{
  const int lane = threadIdx.x & 31;
  const int wave = (blockIdx.x * blockDim.x + threadIdx.x) >> 5;
  const int m    = lane & 15;   // A: channel row / B,D: phase col
  const int half = lane >> 4;

  const int ix = wave & 255;
  const int iy = (wave >> 8) & 127;
  const int b  = wave >> 15;

  // phase of this lane (for B weights and output pixel)
  const int py = m >> 2, px = m & 3;
  const float fyc = 0.25f * (float)py + 0.625f;  // triangle center offset (y)
  const float fxc = 0.25f * (float)px + 0.625f;  // triangle center offset (x)

  const float* sb1 = score + (size_t)(b*CN + m) * (PH*PW);                    // c = m
  const float* sb2 = score + (size_t)(b*CN + 16 + (m < 2 ? m : 2)) * (PH*PW); // c = 16+min(m,2)

  // 2-bit packed corner tables: k -> (cy,cx) = (k/3, k%3) for k<9
  const unsigned CYP = 0x2A540u;
  const unsigned CXP = 0x24924u;

  v8f acc1 = {};  // channels 0..15  x phases 0..15
  v8f acc2 = {};  // channels 16..18 x phases (rows 3..15 = don't-care)

  #pragma unroll
  for (int kc = 0; kc < 3; ++kc) {
    v2f a1, a2, bm;
    #pragma unroll
    for (int s = 0; s < 2; ++s) {
      const int k  = 4*kc + 2*half + s;     // this lane's K-slice (ISA A/B layout)
      const int kk = k < 9 ? k : 8;         // clamp loads; padding zeroed via B
      const int cy = (CYP >> (2*kk)) & 3;
      const int cx = (CXP >> (2*kk)) & 3;
      const int y  = min(max(iy - 1 + cy, 0), PH - 1);
      const int x  = min(max(ix - 1 + cx, 0), PW - 1);
      const int off = y * PW + x;
      const float av1 = sb1[off];
      const float av2 = sb2[off];
      const float wv = fmaxf(0.f, 1.f - fabsf(fyc - (float)cy)) *
                       fmaxf(0.f, 1.f - fabsf(fxc - (float)cx));
      const float bv = (k < 9) ? wv : 0.f;
      if (s == 0) { a1.x = av1; a2.x = av2; bm.x = bv; }
      else        { a1.y = av1; a2.y = av2; bm.y = bv; }
    }
    acc1 = __builtin_amdgcn_wmma_f32_16x16x4_f32(false, a1, false, bm, (short)0, acc1, false, false);
    acc2 = __builtin_amdgcn_wmma_f32_16x16x4_f32(false, a2, false, bm, (short)0, acc2, false, false);
  }

  // lane(m,half) holds channels 8*half+r (acc1) and 16+r (acc2, half==0, r<3)
  float mx = -__builtin_inff();
  #pragma unroll
  for (int r = 0; r < 8; ++r) mx = fmaxf(mx, acc1[r]);
  if (half == 0) { mx = fmaxf(mx, acc2[0]); mx = fmaxf(mx, acc2[1]); mx = fmaxf(mx, acc2[2]); }
  mx = fmaxf(mx, __shfl_xor(mx, 16, 32));

  float ssum = 0.f;
  #pragma unroll
  for (int r = 0; r < 8; ++r) ssum += __expf(acc1[r] - mx);
  if (half == 0) ssum += __expf(acc2[0]-mx) + __expf(acc2[1]-mx) + __expf(acc2[2]-mx);
  ssum += __shfl_xor(ssum, 16, 32);
  const float lse = mx + __logf(ssum);

  const int oy = 4*iy + py, ox = 4*ix + px;
  const int pidx = (b*HH + oy)*WW + ox;
  const int t = target[pidx];
  const bool valid = (t != -2);          // IGNORE_LABEL - 1 = -2
  const int t0 = valid ? t : 0;

  float cand = -__builtin_inff();
  if (t0 < 16) { if ((t0 >> 3) == half) cand = acc1[t0 & 7]; }
  else if (half == 0) cand = acc2[t0 - 16];
  const float pv = fmaxf(cand, __shfl_xor(cand, 16, 32));   // wave-uniform per m

  const float lossv = valid ? (lse - pv) : 0.f;
  const float predm = valid ? pv : __builtin_inff();        // reference pred_masked

  const unsigned bal = (unsigned)__ballot(valid && half == 0);
  if (half == 0) {
    pred[pidx] = predm;
    atomicAdd(&hist[orderKey(predm) >> 16], 1u);
    if (lane == 0) atomicAdd(&meta[0], (unsigned)__popc(bal));
  } else {
    loss[pidx] = lossv;
  }
}

// ---------------------------------------------------------------------------
// Kernel B: find bin of high 16 bits containing the k-th smallest key.
// ---------------------------------------------------------------------------
__global__ __launch_bounds__(256) void select_hi(const unsigned* __restrict__ hist,
                                                 unsigned* __restrict__ meta)
{
  __shared__ unsigned cs[256];
  const int tid = threadIdx.x;
  unsigned s = 0;
  for (int j = 0; j < 256; ++j) s += hist[(tid << 8) | j];
  cs[tid] = s;
  __syncthreads();
  if (tid == 0) {
    const unsigned nv = meta[0];
    unsigned kq = nv > 0 ? nv - 1 : 0;
    if (kq > MIN_KEPT_K) kq = MIN_KEPT_K;
    unsigned cum = 0; int ch = 255;
    for (int i = 0; i < 256; ++i) { if (cum + cs[i] > kq) { ch = i; break; } cum += cs[i]; }
    unsigned c2 = cum; int bin = 255;
    for (int j = 0; j < 256; ++j) {
      const unsigned h = hist[(ch << 8) | j];
      if (c2 + h > kq) { bin = j; break; }
      c2 += h;
    }
    meta[1] = (unsigned)((ch << 8) | bin);  // high-16 bin
    meta[2] = kq - c2;                      // residual rank inside that bin
  }
}

// ---------------------------------------------------------------------------
// Kernel C: histogram of low 16 bits restricted to the selected high bin.
// ---------------------------------------------------------------------------
__global__ __launch_bounds__(256) void hist_low(const float* __restrict__ pred,
                                                const unsigned* __restrict__ meta,
                                                unsigned* __restrict__ hist2)
{
  const unsigned hi = meta[1];
  const float4* p4 = (const float4*)pred;
  for (int i = blockIdx.x * blockDim.x + threadIdx.x; i < NPIX/4; i += gridDim.x * blockDim.x) {
    const float4 v = p4[i];
    unsigned k0 = orderKey(v.x), k1 = orderKey(v.y), k2 = orderKey(v.z), k3 = orderKey(v.w);
    if ((k0 >> 16) == hi) atomicAdd(&hist2[k0 & 0xFFFFu], 1u);
    if ((k1 >> 16) == hi) atomicAdd(&hist2[k1 & 0xFFFFu], 1u);
    if ((k2 >> 16) == hi) atomicAdd(&hist2[k2 & 0xFFFFu], 1u);
    if ((k3 >> 16) == hi) atomicAdd(&hist2[k3 & 0xFFFFu], 1u);
  }
}

// ---------------------------------------------------------------------------
// Kernel D: find low 16 bits -> exact k-th value -> threshold = max(v, 0.7).
// ---------------------------------------------------------------------------
__global__ __launch_bounds__(256) void select_lo(const unsigned* __restrict__ hist2,
                                                 unsigned* __restrict__ meta)
{
  __shared__ unsigned cs[256];
  const int tid = threadIdx.x;
  unsigned s = 0;
  for (int j = 0; j < 256; ++j) s += hist2[(tid << 8) | j];
  cs[tid] = s;
  __syncthreads();
  if (tid == 0) {
    const unsigned rank = meta[2];
    unsigned cum = 0; int ch = 255;
    for (int i = 0; i < 256; ++i) { if (cum + cs[i] > rank) { ch = i; break; } cum += cs[i]; }
    unsigned c2 = cum; int bin = 255;
    for (int j = 0; j < 256; ++j) {
      const unsigned h = hist2[(ch << 8) | j];
      if (c2 + h > rank) { bin = j; break; }
      c2 += h;
    }
    const unsigned key = (meta[1] << 16) | (unsigned)((ch << 8) | bin);
    const float thr = fmaxf(keyToFloat(key), 0.7f);
    meta[3] = __float_as_uint(thr);
  }
}

// ---------------------------------------------------------------------------
// Kernel E: keep = pred < threshold; deterministic block partial sums.
// ---------------------------------------------------------------------------
__global__ __launch_bounds__(256) void ohem_keep(const float* __restrict__ pred,
                                                 const float* __restrict__ loss,
                                                 const unsigned* __restrict__ meta,
                                                 float* __restrict__ partials)
{
  const float thr = __uint_as_float(meta[3]);
  const float4* p4 = (const float4*)pred;
  const float4* l4 = (const float4*)loss;
  float s = 0.f, c = 0.f;
  for (int i = blockIdx.x * blockDim.x + threadIdx.x; i < NPIX/4; i += gridDim.x * blockDim.x) {
    const float4 p = p4[i];
    const float4 l = l4[i];
    if (p.x < thr) { s += l.x; c += 1.f; }   // invalid stored as +inf -> excluded
    if (p.y < thr) { s += l.y; c += 1.f; }
    if (p.z < thr) { s += l.z; c += 1.f; }
    if (p.w < thr) { s += l.w; c += 1.f; }
  }
  __shared__ float ss[256], sc[256];
  const int tid = threadIdx.x;
  ss[tid] = s; sc[tid] = c;
  __syncthreads();
  for (int off = 128; off > 0; off >>= 1) {
    if (tid < off) { ss[tid] += ss[tid + off]; sc[tid] += sc[tid + off]; }
    __syncthreads();
  }
  if (tid == 0) { partials[blockIdx.x] = ss[0]; partials[GRID_RED + blockIdx.x] = sc[0]; }
}

// ---------------------------------------------------------------------------
// Kernel F: final deterministic reduction + division.
// ---------------------------------------------------------------------------
__global__ __launch_bounds__(256) void finalize(const float* __restrict__ partials,
                                                float* __restrict__ out)
{
  __shared__ float ss[256], sc[256];
  const int tid = threadIdx.x;
  float s = 0.f, c = 0.f;
  for (int i = tid; i < GRID_RED; i += 256) { s += partials[i]; c += partials[GRID_RED + i]; }
  ss[tid] = s; sc[tid] = c;
  __syncthreads();
  for (int off = 128; off > 0; off >>= 1) {
    if (tid < off) { ss[tid] += ss[tid + off]; sc[tid] += sc[tid + off]; }
    __syncthreads();
  }
  if (tid == 0) out[0] = ss[0] / fmaxf(sc[0], 1.f);
}

extern "C" void kernel_launch(void* const* d_in, const int* in_sizes, int n_in,
                              void* d_out, int out_size, void* d_ws, size_t ws_size,
                              hipStream_t stream) {
  const float* score  = (const float*)d_in[0];
  const int*   target = (const int*)d_in[1];
  float* out = (float*)d_out;

  float*    predW = (float*)d_ws;                 // NPIX
  float*    lossW = predW + NPIX;                 // NPIX
  unsigned* hist  = (unsigned*)(lossW + NPIX);    // 65536
  unsigned* hist2 = hist + 65536;                 // 65536
  unsigned* meta  = hist2 + 65536;                // 16
  float* partials = (float*)(meta + 16);          // 2 * GRID_RED

  // zero hist, hist2, meta each call (graph-capture-safe async memset)
  hipMemsetAsync(hist, 0, (2u * 65536u + 16u) * sizeof(unsigned), stream);

  // 8*128*256 tiles, one wave32 per tile, 8 waves per block
  ohem_interp_ce<<<(BN * PH * PW) / 8, 256, 0, stream>>>(score, target, predW, lossW, hist, meta);
  select_hi<<<1, 256, 0, stream>>>(hist, meta);
  hist_low<<<GRID_RED, 256, 0, stream>>>(predW, meta, hist2);
  select_lo<<<1, 256, 0, stream>>>(hist2, meta);
  ohem_keep<<<GRID_RED, 256, 0, stream>>>(predW, lossW, meta, partials);
  finalize<<<1, 256, 0, stream>>>(partials, out);
}